// DistToClosest_39470749450747
// MI455X (gfx1250) — compile-verified
//
#include <hip/hip_runtime.h>

typedef __attribute__((ext_vector_type(2))) float v2f;
typedef __attribute__((ext_vector_type(8))) float v8f;

#define YSPLIT 4

// ---------------------------------------------------------------------------
// Pass 1: pack B operand tiles for V_WMMA_F32_16X16X4_F32.
// B is 4x16 (KxN) per tile, 2 VGPRs/lane in the ISA layout:
//   VGPR0: lanes 0-15 -> K=0, lanes 16-31 -> K=2
//   VGPR1: lanes 0-15 -> K=1, lanes 16-31 -> K=3
// We store per tile 32 lanes x 2 floats contiguously so the hot loop does one
// coalesced global_load_b64 per lane:  bpack[t*64 + lane*2 + {0,1}].
// B column j = [y_j0, y_j1, y_j2, ||y_j||^2]  (folds the y^2 term into the GEMM)
// ---------------------------------------------------------------------------
__global__ void pack_b_kernel(const float* __restrict__ y,
                              float* __restrict__ bpack, int ytiles) {
  int tid = blockIdx.x * blockDim.x + threadIdx.x;
  if (tid >= ytiles * 32) return;
  int t = tid >> 5;
  int lane = tid & 31;
  int j = t * 16 + (lane & 15);
  float y0 = y[j * 3 + 0];
  float y1 = y[j * 3 + 1];
  float y2 = y[j * 3 + 2];
  float o0, o1;
  if (lane < 16) {        // K=0, K=1
    o0 = y0; o1 = y1;
  } else {                // K=2, K=3
    o0 = y2; o1 = y0 * y0 + y1 * y1 + y2 * y2;
  }
  bpack[(size_t)t * 64 + lane * 2 + 0] = o0;
  bpack[(size_t)t * 64 + lane * 2 + 1] = o1;
}

// ---------------------------------------------------------------------------
// Pass 2: per wave, one 16-row x-tile and 1/YSPLIT of the y tiles.
// A row i = [-2*x_i0, -2*x_i1, -2*x_i2, 1.0]; A layout (16x4 f32):
//   VGPR0: lanes 0-15 -> K=0, lanes 16-31 -> K=2
//   VGPR1: lanes 0-15 -> K=1, lanes 16-31 -> K=3     (M = lane % 16)
// Each tile: 1 x v_wmma_f32_16x16x4_f32 (C=0) giving D_ij = -2 x.y + y2_j,
// then 8 running v_min_f32. Cross-lane min deferred to after the loop.
// C/D layout: VGPR v, lanes 0-15: row M=v; lanes 16-31: row M=v+8.
// ---------------------------------------------------------------------------
__global__ __launch_bounds__(128) void chamfer_wmma_kernel(
    const float* __restrict__ x, const float* __restrict__ bpack,
    float* __restrict__ partial, int n, int ytiles) {
  const int lane  = threadIdx.x & 31;
  const int gwave = blockIdx.x * (blockDim.x >> 5) + (threadIdx.x >> 5);
  const int xtile  = gwave >> 2;      // /YSPLIT
  const int ysplit = gwave & (YSPLIT - 1);

  // Build A operand (lives in 2 VGPRs for the whole loop).
  const int m   = lane & 15;
  const int row = xtile * 16 + m;
  v2f A;
  if (lane < 16) {
    A.x = -2.0f * x[row * 3 + 0];
    A.y = -2.0f * x[row * 3 + 1];
  } else {
    A.x = -2.0f * x[row * 3 + 2];
    A.y = 1.0f;
  }

  v8f mn;
#pragma unroll
  for (int v = 0; v < 8; ++v) mn[v] = __builtin_inff();

  const int tiles_per = ytiles / YSPLIT;
  const float* bp = bpack + (size_t)(ysplit * tiles_per) * 64 + lane * 2;

#pragma unroll 4
  for (int t = 0; t < tiles_per; ++t) {
    v2f B = *(const v2f*)(bp + (size_t)t * 64);
    v8f c = {};
    // D = A x B + 0  ->  v_wmma_f32_16x16x4_f32
    c = __builtin_amdgcn_wmma_f32_16x16x4_f32(
        /*neg_a=*/false, A, /*neg_b=*/false, B,
        /*c_mod=*/(short)0, c, /*reuse_a=*/false, /*reuse_b=*/false);
#pragma unroll
    for (int v = 0; v < 8; ++v) mn[v] = fminf(mn[v], c[v]);
  }

  // Cross-lane min over the 16 N-lanes (xor 8,4,2,1 stays within each half).
#pragma unroll
  for (int off = 8; off >= 1; off >>= 1) {
#pragma unroll
    for (int v = 0; v < 8; ++v) {
      float o = __shfl_xor(mn[v], off, 32);
      mn[v] = fminf(mn[v], o);
    }
  }

  // Lane 0 holds rows 0..7 of the tile, lane 16 holds rows 8..15.
  float* dst = partial + (size_t)ysplit * n + xtile * 16;
  if (lane == 0) {
#pragma unroll
    for (int v = 0; v < 8; ++v) dst[v] = mn[v];
  } else if (lane == 16) {
#pragma unroll
    for (int v = 0; v < 8; ++v) dst[8 + v] = mn[v];
  }
}

// ---------------------------------------------------------------------------
// Pass 3: merge the YSPLIT partial mins, add ||x_i||^2, write closest_dists
// to d_out[1 + i].
// ---------------------------------------------------------------------------
__global__ void finalize_kernel(const float* __restrict__ x,
                                const float* __restrict__ partial,
                                float* __restrict__ out, int n) {
  int i = blockIdx.x * blockDim.x + threadIdx.x;
  if (i >= n) return;
  float mv = partial[i];
#pragma unroll
  for (int s = 1; s < YSPLIT; ++s) mv = fminf(mv, partial[(size_t)s * n + i]);
  float x0 = x[i * 3 + 0], x1 = x[i * 3 + 1], x2 = x[i * 3 + 2];
  out[1 + i] = mv + (x0 * x0 + x1 * x1 + x2 * x2);
}

// ---------------------------------------------------------------------------
// Pass 4: deterministic sum of closest_dists into d_out[0] (single block,
// fixed-order strided accumulation + LDS tree; no float atomics).
// ---------------------------------------------------------------------------
__global__ void sum_kernel(const float* __restrict__ vals,
                           float* __restrict__ out0, int n) {
  __shared__ float s[256];
  float acc = 0.0f;
  for (int i = threadIdx.x; i < n; i += 256) acc += vals[i];
  s[threadIdx.x] = acc;
  __syncthreads();
  for (int stride = 128; stride >= 1; stride >>= 1) {
    if ((int)threadIdx.x < stride) s[threadIdx.x] += s[threadIdx.x + stride];
    __syncthreads();
  }
  if (threadIdx.x == 0) *out0 = s[0];
}

extern "C" void kernel_launch(void* const* d_in, const int* in_sizes, int n_in,
                              void* d_out, int out_size, void* d_ws, size_t ws_size,
                              hipStream_t stream) {
  (void)n_in; (void)out_size; (void)ws_size;
  const float* x = (const float*)d_in[0];
  const float* y = (const float*)d_in[1];
  float* out = (float*)d_out;

  const int n = in_sizes[0] / 3;   // 16384
  const int m = in_sizes[1] / 3;   // 16384
  const int xtiles = n / 16;       // 1024
  const int ytiles = m / 16;       // 1024

  float* bpack   = (float*)d_ws;                       // ytiles*64 floats (256 KB)
  float* partial = bpack + (size_t)ytiles * 64;        // YSPLIT*n floats (256 KB)

  pack_b_kernel<<<(ytiles * 32 + 255) / 256, 256, 0, stream>>>(y, bpack, ytiles);

  // 4 waves per block; total waves = xtiles * YSPLIT.
  chamfer_wmma_kernel<<<xtiles * YSPLIT / 4, 128, 0, stream>>>(x, bpack, partial,
                                                               n, ytiles);

  finalize_kernel<<<(n + 255) / 256, 256, 0, stream>>>(x, partial, out, n);
  sum_kernel<<<1, 256, 0, stream>>>(out + 1, out, n);
}